// QuantumLayer_31636729102603
// MI455X (gfx1250) — compile-verified
//
#include <hip/hip_runtime.h>
#include <hip/hip_bf16.h>

#define TPB 256

// ---------------------------------------------------------------------------
// Kernel 1: fold q_weights[3][2][3] into the 3x3 bilinear-form matrix R.
//   out(x0,x1) = (1,cos x0,sin x0) . R . (1,cos x1,sin x1)^T
// Runs on a single lane; cost is negligible (a few hundred scalar flops).
// ---------------------------------------------------------------------------
__global__ void qexp_setup(const float* __restrict__ w, float* __restrict__ R9) {
    if (threadIdx.x != 0 || blockIdx.x != 0) return;

    float Mr[4][4], Mi[4][4];
    for (int i = 0; i < 4; ++i)
        for (int j = 0; j < 4; ++j) { Mr[i][j] = (i == j) ? 1.f : 0.f; Mi[i][j] = 0.f; }

    // CNOT(0,1) then CNOT(1,0) as a row permutation of (U0 (x) U1)
    const int perm[4] = {0, 2, 3, 1};

    for (int l = 0; l < 3; ++l) {
        float Ur[2][2][2], Ui[2][2][2];              // [qubit][row][col]
        for (int q = 0; q < 2; ++q) {
            float phi = w[l * 6 + q * 3 + 0];
            float th  = w[l * 6 + q * 3 + 1];
            float om  = w[l * 6 + q * 3 + 2];
            float ct = cosf(0.5f * th), st = sinf(0.5f * th);
            float ap = -0.5f * (phi + om);           // ep = e^{i*ap}
            float am =  0.5f * (phi - om);           // em = e^{i*am}
            float epr = cosf(ap), epi = sinf(ap);
            float emr = cosf(am), emi = sinf(am);
            Ur[q][0][0] =  epr * ct;  Ui[q][0][0] =  epi * ct;
            Ur[q][0][1] = -emr * st;  Ui[q][0][1] = -emi * st;
            Ur[q][1][0] =  emr * st;  Ui[q][1][0] = -emi * st;   // conj(em)*s
            Ur[q][1][1] =  epr * ct;  Ui[q][1][1] = -epi * ct;   // conj(ep)*c
        }
        float Gr[4][4], Gi[4][4];                    // G = U0 (x) U1
        for (int i0 = 0; i0 < 2; ++i0)
            for (int i1 = 0; i1 < 2; ++i1)
                for (int j0 = 0; j0 < 2; ++j0)
                    for (int j1 = 0; j1 < 2; ++j1) {
                        float ar = Ur[0][i0][j0], ai = Ui[0][i0][j0];
                        float br = Ur[1][i1][j1], bi = Ui[1][i1][j1];
                        Gr[2 * i0 + i1][2 * j0 + j1] = ar * br - ai * bi;
                        Gi[2 * i0 + i1][2 * j0 + j1] = ar * bi + ai * br;
                    }
        float Nr[4][4], Ni[4][4];                    // M <- (perm-rows of G) * M
        for (int i = 0; i < 4; ++i)
            for (int j = 0; j < 4; ++j) {
                float sr = 0.f, si = 0.f;
                for (int t = 0; t < 4; ++t) {
                    float lr = Gr[perm[i]][t], li = Gi[perm[i]][t];
                    sr += lr * Mr[t][j] - li * Mi[t][j];
                    si += lr * Mi[t][j] + li * Mr[t][j];
                }
                Nr[i][j] = sr; Ni[i][j] = si;
            }
        for (int i = 0; i < 4; ++i)
            for (int j = 0; j < 4; ++j) { Mr[i][j] = Nr[i][j]; Mi[i][j] = Ni[i][j]; }
    }

    // Initial state a = (v0, -i v1, -i v2, -v3), v = (c0c1, c0s1, s0c1, s0s1).
    // out = sum_k s_k |(M a)_k|^2 = v^T Q v, Q symmetric.
    float Q[4][4] = {};
    for (int k = 0; k < 4; ++k) {
        float sk = (k < 2) ? 1.f : -1.f;             // <Z0> sign per basis state
        float ar[4] = {  Mr[k][0],  Mi[k][1],  Mi[k][2], -Mr[k][3] };
        float ai[4] = {  Mi[k][0], -Mr[k][1], -Mr[k][2], -Mi[k][3] };
        for (int i = 0; i < 4; ++i)
            for (int j = 0; j < 4; ++j)
                Q[i][j] += sk * (ar[i] * ar[j] + ai[i] * ai[j]);
    }

    // Half-angle products -> full-angle basis (1, cos x, sin x):
    //   c*c = (1+C)/2, c*s = S/2, s*s = (1-C)/2
    const float A[3][3] = {{0.5f, 0.5f, 0.f}, {0.f, 0.f, 0.5f}, {0.5f, -0.5f, 0.f}};
    float R[3][3] = {};
    for (int i = 0; i < 4; ++i)
        for (int j = 0; j < 4; ++j) {
            int t0 = (i >> 1) + (j >> 1);            // 0:c0c0 1:c0s0 2:s0s0
            int t1 = (i & 1) + (j & 1);              // same for qubit 1
            for (int a = 0; a < 3; ++a)
                for (int b = 0; b < 3; ++b)
                    R[a][b] += Q[i][j] * A[t0][a] * A[t1][b];
        }
    for (int a = 0; a < 3; ++a)
        for (int b = 0; b < 3; ++b) R9[a * 3 + b] = R[a][b];
}

// ---------------------------------------------------------------------------
// Kernel 2: streaming evaluation. 4 elements/thread, inputs staged through
// LDS with the CDNA5 async engine (global_load_async_to_lds_b128 +
// s_wait_asynccnt). Each lane reads only its own 32B LDS slot -> wave-level
// wait is sufficient, no workgroup barrier.
// ---------------------------------------------------------------------------
__device__ __forceinline__ float evalElem(float x0, float x1,
                                          float4 q0, float4 q1, float q2) {
    float s0, c0, s1, c1;
    __sincosf(x0, &s0, &c0);
    __sincosf(x1, &s1, &c1);
    // R row-major: q0=(R00,R01,R02,R10) q1=(R11,R12,R20,R21) q2=R22
    float t0 = fmaf(q1.z, s0, fmaf(q0.w, c0, q0.x));
    float t1 = fmaf(q1.w, s0, fmaf(q1.x, c0, q0.y));
    float t2 = fmaf(q2,   s0, fmaf(q1.y, c0, q0.z));
    return fmaf(t2, s1, fmaf(t1, c1, t0));
}

__global__ __launch_bounds__(TPB) void qexp_main(const float* __restrict__ x,
                                                 const float* __restrict__ R9,
                                                 float* __restrict__ out, int B) {
    __shared__ alignas(16) float tile[TPB * 8];      // 8 KB: two 4 KB regions
    const int tid = threadIdx.x;
    const int blockStart = blockIdx.x * (TPB * 4);   // 1024 elements per block
    const int eA = blockStart + 2 * tid;             // region A: first 512 elems
    const int eB = blockStart + 2 * TPB + 2 * tid;   // region B: next 512 elems

    const bool fullA = (eA + 2) <= B;
    const bool fullB = (eB + 2) <= B;

    // Low 32 bits of a generic shared pointer == wave-relative LDS byte offset.
    unsigned ldsA = (unsigned)(reinterpret_cast<size_t>(&tile[4 * tid]));
    unsigned ldsB = (unsigned)(reinterpret_cast<size_t>(&tile[4 * TPB + 4 * tid]));

    if (fullA) {
        unsigned long long ga =
            (unsigned long long)(reinterpret_cast<size_t>(x + 2 * eA));
        asm volatile("global_load_async_to_lds_b128 %0, %1, off"
                     :: "v"(ldsA), "v"(ga) : "memory");
    }
    if (fullB) {
        unsigned long long gb =
            (unsigned long long)(reinterpret_cast<size_t>(x + 2 * eB));
        asm volatile("global_load_async_to_lds_b128 %0, %1, off"
                     :: "v"(ldsB), "v"(gb) : "memory");
    }

    // Overlap the (uniform, L2-hot) coefficient loads with the async DMA.
    const float4 q0 = *reinterpret_cast<const float4*>(R9);
    const float4 q1 = *reinterpret_cast<const float4*>(R9 + 4);
    const float  q2 = R9[8];

    asm volatile("s_wait_asynccnt 0" ::: "memory");

    if (fullA) {
        float4 v = *reinterpret_cast<const float4*>(&tile[4 * tid]);
        float2 r;
        r.x = evalElem(v.x, v.y, q0, q1, q2);
        r.y = evalElem(v.z, v.w, q0, q1, q2);
        *reinterpret_cast<float2*>(out + eA) = r;
    } else if (eA < B) {
        for (int e = eA; e < B && e < eA + 2; ++e)
            out[e] = evalElem(x[2 * e], x[2 * e + 1], q0, q1, q2);
    }
    if (fullB) {
        float4 v = *reinterpret_cast<const float4*>(&tile[4 * TPB + 4 * tid]);
        float2 r;
        r.x = evalElem(v.x, v.y, q0, q1, q2);
        r.y = evalElem(v.z, v.w, q0, q1, q2);
        *reinterpret_cast<float2*>(out + eB) = r;
    } else if (eB < B) {
        for (int e = eB; e < B && e < eB + 2; ++e)
            out[e] = evalElem(x[2 * e], x[2 * e + 1], q0, q1, q2);
    }
}

extern "C" void kernel_launch(void* const* d_in, const int* in_sizes, int n_in,
                              void* d_out, int out_size, void* d_ws, size_t ws_size,
                              hipStream_t stream) {
    (void)n_in; (void)out_size; (void)ws_size;
    const float* x = (const float*)d_in[0];      // [B,2] float32
    const float* w = (const float*)d_in[1];      // [3,2,3] float32
    float* out = (float*)d_out;                  // [B] float32
    float* R9  = (float*)d_ws;                   // 9-float scratch

    const int B = in_sizes[0] / 2;
    if (B <= 0) return;

    qexp_setup<<<1, 64, 0, stream>>>(w, R9);
    const int blocks = (B + TPB * 4 - 1) / (TPB * 4);
    qexp_main<<<blocks, TPB, 0, stream>>>(x, R9, out, B);
}